// LIF_17746804867228
// MI455X (gfx1250) — compile-verified
//
#include <hip/hip_runtime.h>

// LIF membrane scan: x[T,B,D] f32 -> spikes[T,B,D] f32.
// T=64 sequential steps; B*D independent lanes carried in registers.
// Memory-bound (AI ~0.5 flop/byte): optimize the gfx1250 memory path only.
//   - b128 vector loads/stores (4 floats / thread / step)
//   - non-temporal hints: single-pass 1 GB stream must not pollute 192MB L2
//   - global_prefetch_b8 of the next timestep slab (no LOADcnt cost)
//   - unroll 4 => up to 4 independent global_load_b128 in flight per wave

typedef __attribute__((ext_vector_type(4))) float v4f;

#define LIF_T      64
#define LIF_TAU    1.0f
#define LIF_THRESH 1.0f

__global__ __launch_bounds__(256) void lif_scan_kernel(
    const float* __restrict__ x, float* __restrict__ out, int n_vec) {
  const int i = blockIdx.x * blockDim.x + threadIdx.x;
  if (i >= n_vec) return;

  const v4f* __restrict__ xp = reinterpret_cast<const v4f*>(x) + i;
  v4f* __restrict__ op = reinterpret_cast<v4f*>(out) + i;
  const size_t stride = (size_t)n_vec;  // float4s per timestep

  v4f mem = {0.0f, 0.0f, 0.0f, 0.0f};

#pragma unroll 4
  for (int t = 0; t < LIF_T; ++t) {
    // Streaming, single-use data: non-temporal 128-bit load.
    v4f v = __builtin_nontemporal_load(xp);

    // Prefetch next timestep's cacheline (gfx1250 global_prefetch_b8;
    // does not touch LOADcnt). Guarded so we never prefetch past the buffer.
    if (t + 1 < LIF_T) {
      __builtin_prefetch((const void*)(xp + stride), 0, 3);
    }

    // mem = tau*mem + x_t
    mem = mem * LIF_TAU + v;

    // spike = (mem - thresh >= 0) * thresh ; hard reset mem *= (1 - spike)
    v4f sp;
    sp.x = (mem.x >= LIF_THRESH) ? LIF_THRESH : 0.0f;
    sp.y = (mem.y >= LIF_THRESH) ? LIF_THRESH : 0.0f;
    sp.z = (mem.z >= LIF_THRESH) ? LIF_THRESH : 0.0f;
    sp.w = (mem.w >= LIF_THRESH) ? LIF_THRESH : 0.0f;

    // Output is write-once: non-temporal 128-bit store.
    __builtin_nontemporal_store(sp, op);

    v4f one = {1.0f, 1.0f, 1.0f, 1.0f};
    mem = mem * (one - sp);

    xp += stride;
    op += stride;
  }
}

extern "C" void kernel_launch(void* const* d_in, const int* in_sizes, int n_in,
                              void* d_out, int out_size, void* d_ws, size_t ws_size,
                              hipStream_t stream) {
  const float* x = (const float*)d_in[0];
  float* out = (float*)d_out;

  const int total = in_sizes[0];      // 64*128*16384 = 134,217,728
  const int N = total / LIF_T;        // elements per timestep (B*D)
  const int n_vec = N / 4;            // float4 lanes (D divisible by 4)

  const int block = 256;              // 8 wave32s
  const int grid = (n_vec + block - 1) / block;  // 2048 blocks

  lif_scan_kernel<<<grid, block, 0, stream>>>(x, out, n_vec);
}